// mLSTM_86105504350522
// MI455X (gfx1250) — compile-verified
//
#include <hip/hip_runtime.h>
#include <math.h>

#define D 4096
#define INV_SQRT_MD 0.015625f  // 1/sqrt(4096)

typedef __attribute__((ext_vector_type(2))) float v2f;
typedef __attribute__((ext_vector_type(8))) float v8f;

// ---------------------------------------------------------------------------
// Kernel 1: four GEMVs (W_q, W_k, W_v, W_o) @ x via V_WMMA_F32_16X16X4_F32.
// One wave = 16 output rows of one matrix. 1024 waves total.
// A layout (f32 16x4): lane<16 -> row r0+lane, K={0,1}; lane>=16 -> row, K={2,3}
// B replicated across all 16 columns: B[k][n] = x[k0+k]  -> every D column = W@x
// ---------------------------------------------------------------------------
__global__ __launch_bounds__(256) void gemv4_wmma(
    const float* __restrict__ x,
    const float* __restrict__ Wq, const float* __restrict__ bq,
    const float* __restrict__ Wk, const float* __restrict__ bk,
    const float* __restrict__ Wv, const float* __restrict__ bv,
    const float* __restrict__ Wo, const float* __restrict__ bo,
    float* __restrict__ ws_qkvo)            // [4][D]: q, k_scaled, v, o_pre
{
    const int tid   = threadIdx.x;
    const int lane  = tid & 31;
    const int gwave = blockIdx.x * 8 + (tid >> 5);   // 0..1023
    const int mat   = gwave >> 8;                    // 0..3
    const int r0    = (gwave & 255) * 16;

    const float* W; const float* bias; float scale;
    switch (mat) {
      case 0:  W = Wq; bias = bq; scale = 1.0f;        break;
      case 1:  W = Wk; bias = bk; scale = INV_SQRT_MD; break;
      case 2:  W = Wv; bias = bv; scale = 1.0f;        break;
      default: W = Wo; bias = bo; scale = 1.0f;        break;
    }
    float* out = ws_qkvo + mat * D;

    const int arow = r0 + (lane & 15);
    const int koff = (lane < 16) ? 0 : 2;            // which K pair this lane holds
    const float* wrow = W + (size_t)arow * D + koff;

    v8f acc0 = {}; v8f acc1 = {};
    for (int k0 = 0; k0 < D; k0 += 8) {
        v2f a0 = *(const v2f*)(wrow + k0);
        v2f a1 = *(const v2f*)(wrow + k0 + 4);
        v2f b0, b1;
        b0.x = x[k0 + koff];     b0.y = x[k0 + koff + 1];
        b1.x = x[k0 + 4 + koff]; b1.y = x[k0 + 4 + koff + 1];
        acc0 = __builtin_amdgcn_wmma_f32_16x16x4_f32(false, a0, false, b0,
                                                     (short)0, acc0, false, false);
        acc1 = __builtin_amdgcn_wmma_f32_16x16x4_f32(false, a1, false, b1,
                                                     (short)0, acc1, false, false);
    }
    v8f acc = acc0 + acc1;
    // C/D layout: lane n holds column n; VGPR j = row r0+j (lanes 0-15) / r0+8+j (16-31)
    if (lane == 0) {
        for (int j = 0; j < 8; ++j) out[r0 + j]     = (acc[j] + bias[r0 + j])     * scale;
    } else if (lane == 16) {
        for (int j = 0; j < 8; ++j) out[r0 + 8 + j] = (acc[j] + bias[r0 + 8 + j]) * scale;
    }
}

// ---------------------------------------------------------------------------
// Kernel 2 (1 block): gate scalars i_t, f_t; n = f*n_prev + i*k; max(|n.q|,1)
// ---------------------------------------------------------------------------
__global__ __launch_bounds__(256) void gates_n_kernel(
    const float* __restrict__ x,
    const float* __restrict__ wi, const float* __restrict__ bi,
    const float* __restrict__ wf, const float* __restrict__ bf,
    const float* __restrict__ n_prev,
    const float* __restrict__ ws_qkvo,
    float* __restrict__ scalars,            // [0]=i_t [1]=f_t [2]=1/max_nqt
    float* __restrict__ n_out)
{
    __shared__ float red[256];
    const int tid = threadIdx.x;
    const float* q  = ws_qkvo;
    const float* ks = ws_qkvo + D;

    float di = 0.f, df = 0.f;
    for (int r = tid; r < D; r += 256) { float xv = x[r]; di += wi[r] * xv; df += wf[r] * xv; }
    red[tid] = di; __syncthreads();
    for (int s = 128; s > 0; s >>= 1) { if (tid < s) red[tid] += red[tid + s]; __syncthreads(); }
    float dot_i = red[0]; __syncthreads();
    red[tid] = df; __syncthreads();
    for (int s = 128; s > 0; s >>= 1) { if (tid < s) red[tid] += red[tid + s]; __syncthreads(); }
    float dot_f = red[0]; __syncthreads();

    float i_t = expf(dot_i + bi[0]);
    float f_t = 1.f / (1.f + expf(-(dot_f + bf[0])));

    float dn = 0.f;
    for (int r = tid; r < D; r += 256) {
        float nr = f_t * n_prev[r] + i_t * ks[r];
        n_out[r] = nr;
        dn += nr * q[r];
    }
    red[tid] = dn; __syncthreads();
    for (int s = 128; s > 0; s >>= 1) { if (tid < s) red[tid] += red[tid + s]; __syncthreads(); }
    if (tid == 0) {
        float m = fabsf(red[0]); if (m < 1.f) m = 1.f;
        scalars[0] = i_t; scalars[1] = f_t; scalars[2] = 1.f / m;
    }
}

// ---------------------------------------------------------------------------
// Kernel 3: C = f*C_prev + i*outer(v,k) as rank-1 WMMAs, fused with h = C @ q.
// Block = 4 waves, owns 16 rows of C; each wave covers 1024 columns.
// ---------------------------------------------------------------------------
__global__ __launch_bounds__(128) void cell_update_wmma(
    const float* __restrict__ C_prev,
    const float* __restrict__ ws_qkvo,
    const float* __restrict__ scalars,
    float* __restrict__ C_out,
    float* __restrict__ h_acc)
{
    __shared__ float red[4][16];
    const int tid  = threadIdx.x;
    const int lane = tid & 31;
    const int wave = tid >> 5;                      // 0..3
    const int r0   = blockIdx.x * 16;
    const float i_t = scalars[0];
    const float f_t = scalars[1];
    const float* q  = ws_qkvo;
    const float* ks = ws_qkvo + D;
    const float* v  = ws_qkvo + 2 * D;

    // A: rank-1 along K. Only K=0 nonzero: A[m][0] = v[r0+m]
    v2f a; a.x = (lane < 16) ? v[r0 + (lane & 15)] : 0.f; a.y = 0.f;

    const int    col_in  = lane & 15;
    const size_t rowbase = (size_t)(r0 + ((lane < 16) ? 0 : 8)) * D;

    float hp[8];
    for (int j = 0; j < 8; ++j) hp[j] = 0.f;

    const int c_begin = wave * (D / 4);
    const int c_end   = c_begin + (D / 4);
    for (int c0 = c_begin; c0 < c_end; c0 += 16) {
        const int col = c0 + col_in;
        v2f b; b.x = (lane < 16) ? (i_t * ks[col]) : 0.f; b.y = 0.f;  // B[0][n]=i*k
        v8f cf;
        for (int j = 0; j < 8; ++j)
            cf[j] = f_t * C_prev[rowbase + (size_t)j * D + col];
        v8f d = __builtin_amdgcn_wmma_f32_16x16x4_f32(false, a, false, b,
                                                      (short)0, cf, false, false);
        for (int j = 0; j < 8; ++j)
            C_out[rowbase + (size_t)j * D + col] = d[j];
        const float qv = q[col];
        for (int j = 0; j < 8; ++j) hp[j] += d[j] * qv;
    }
    // reduce row-dots across each 16-lane half (xor masks stay within a half)
    for (int m = 1; m < 16; m <<= 1)
        for (int j = 0; j < 8; ++j)
            hp[j] += __shfl_xor(hp[j], m, 32);
    if (lane == 0)  for (int j = 0; j < 8; ++j) red[wave][j]     = hp[j];
    if (lane == 16) for (int j = 0; j < 8; ++j) red[wave][8 + j] = hp[j];
    __syncthreads();
    if (tid < 16)
        h_acc[r0 + tid] = red[0][tid] + red[1][tid] + red[2][tid] + red[3][tid];
}

// ---------------------------------------------------------------------------
// Kernel 4: h_t = sigmoid(o_pre) * h_acc / max_nqt
// ---------------------------------------------------------------------------
__global__ __launch_bounds__(256) void output_kernel(
    const float* __restrict__ ws_qkvo,
    const float* __restrict__ scalars,
    const float* __restrict__ h_acc,
    float* __restrict__ h_out)
{
    const int r = blockIdx.x * 256 + threadIdx.x;
    if (r < D) {
        const float o = 1.f / (1.f + expf(-ws_qkvo[3 * D + r]));
        h_out[r] = o * h_acc[r] * scalars[2];
    }
}

// ---------------------------------------------------------------------------
extern "C" void kernel_launch(void* const* d_in, const int* in_sizes, int n_in,
                              void* d_out, int out_size, void* d_ws, size_t ws_size,
                              hipStream_t stream) {
    (void)in_sizes; (void)n_in; (void)out_size; (void)ws_size;
    const float* x      = (const float*)d_in[0];
    const float* C_prev = (const float*)d_in[1];
    const float* n_prev = (const float*)d_in[2];
    const float* W_q    = (const float*)d_in[3];
    const float* b_q    = (const float*)d_in[4];
    const float* W_k    = (const float*)d_in[5];
    const float* b_k    = (const float*)d_in[6];
    const float* W_v    = (const float*)d_in[7];
    const float* b_v    = (const float*)d_in[8];
    const float* w_i    = (const float*)d_in[9];
    const float* b_i    = (const float*)d_in[10];
    const float* w_f    = (const float*)d_in[11];
    const float* b_f    = (const float*)d_in[12];
    const float* W_o    = (const float*)d_in[13];
    const float* b_o    = (const float*)d_in[14];

    float* out   = (float*)d_out;
    float* h_out = out;                 // [D]
    float* C_out = out + D;             // [D*D]
    float* n_out = out + D + (size_t)D * D;  // [D]

    float* wsf     = (float*)d_ws;
    float* qkvo    = wsf;               // 4*D
    float* scalars = wsf + 4 * D;       // 3
    float* h_acc   = wsf + 4 * D + 16;  // D

    gemv4_wmma<<<128, 256, 0, stream>>>(x, W_q, b_q, W_k, b_k, W_v, b_v, W_o, b_o, qkvo);
    gates_n_kernel<<<1, 256, 0, stream>>>(x, w_i, b_i, w_f, b_f, n_prev, qkvo, scalars, n_out);
    cell_update_wmma<<<256, 128, 0, stream>>>(C_prev, qkvo, scalars, C_out, h_acc);
    output_kernel<<<16, 256, 0, stream>>>(qkvo, scalars, h_acc, h_out);
}